// ReversePolicy_83502754169167
// MI455X (gfx1250) — compile-verified
//
#include <hip/hip_runtime.h>
#include <hip/hip_bf16.h>
#include <stdint.h>

// Problem constants (match the reference)
#define N_NODES 1024
#define DN      128
#define DE      64
#define E_EDITS 1000000
#define KFEAT   320             // 2*DN + DE
#define KPAD    324             // padded LDS row pitch (324 % 64 == 4 -> conflict-free A reads)
#define NTILES  (E_EDITS / 16)  // 62500, exact
#define BLOCKS  3125            // 2 waves/block -> 6250 waves -> exactly 10 tiles/wave

typedef __attribute__((ext_vector_type(2))) float v2f;
typedef __attribute__((ext_vector_type(8))) float v8f;

// ---------------------------------------------------------------------------
// Edit-score kernel: one wave32 per tile of 16 edits.
//   1) coalesced gather of the 16x320 feature tile into padded LDS
//   2) 80x V_WMMA_F32_16X16X4_F32 -> 16 edits x 4 channels.
//      B column n carries W row (n&3); columns 4..15 of D are duplicates that
//      are never read -> branch-free B-fragment loads (broadcast, no conflict).
//   3) per-edit channel select + feasibility mask, coalesced store
// ---------------------------------------------------------------------------
__global__ __launch_bounds__(64) void edit_scores_kernel(
    const float* __restrict__ h_nodes,
    const float* __restrict__ h_edges,
    const float* __restrict__ W_edit,
    const float* __restrict__ b_edit,
    const int*   __restrict__ i_idx,
    const int*   __restrict__ j_idx,
    const int*   __restrict__ b_idx,
    const uint8_t* __restrict__ feasible,
    float* __restrict__ out)
{
    __shared__ float ldsW[4 * KPAD];          // W_edit, row pitch padded
    __shared__ float ldsBias[4];
    __shared__ float ldsFeat[2][16 * KPAD];   // per-wave 16x320 feature tile (padded)
    __shared__ float ldsScore[2][16 * 16];    // per-wave D-matrix spill

    const int tid  = threadIdx.x;
    const int lane = tid & 31;
    const int wave = tid >> 5;

    // Cache the tiny scoring head in LDS (once per block).
    for (int t = tid; t < 4 * KFEAT; t += 64) {
        const int r = t / KFEAT;
        const int c = t - r * KFEAT;
        ldsW[r * KPAD + c] = W_edit[t];
    }
    if (tid < 4) ldsBias[tid] = b_edit[tid];
    __syncthreads();

    float* featT  = ldsFeat[wave];
    float* scoreT = ldsScore[wave];

    // B-fragment lane mapping for V_WMMA_F32_16X16X4_F32 (B is 4x16, K x N):
    //   lane holds {B[kOff][nCol], B[kOff+1][nCol]}, kOff = (lane<16)?0:2.
    //   B[:,n] := W_edit[n&3] -> dead D columns hold finite duplicates.
    const int nCol = lane & 15;
    const int kOff = (lane >> 4) * 2;
    const int wRow = nCol & 3;

    // A-fragment mapping (A is 16x4, M x K): lane holds feat[arow][k0+kOff .. +1]
    const int arow = lane & 15;

    const int waveGlobal = blockIdx.x * 2 + wave;
    const int waveCount  = gridDim.x * 2;

    for (int tile = waveGlobal; tile < NTILES; tile += waveCount) {
        const int base = tile * 16;

        // Lanes 0..15 own one edit each: fetch indices (coalesced 64B per array).
        int iv = 0, jv = 0, bv = 0, fv = 0;
        if (lane < 16) {
            iv = i_idx[base + lane];
            jv = j_idx[base + lane];
            bv = b_idx[base + lane];
            fv = (int)feasible[base + lane];
        }

        // Prefetch both 128B cache lines of every edge row in this tile
        // (gfx1250 global_prefetch_b8) before the gather loop touches them.
        if (lane < 16) {
            const char* ep = (const char*)(h_edges +
                ((size_t)iv * N_NODES + (size_t)jv) * DE);
            __builtin_prefetch(ep,       0, 0);
            __builtin_prefetch(ep + 128, 0, 0);
        }

        // Coalesced gather of the 16x320 tile into LDS.
        for (int m = 0; m < 16; ++m) {
            const int im = __shfl(iv, m, 32);
            const int jm = __shfl(jv, m, 32);
            float* row = featT + m * KPAD;

            // h_nodes[im]: 128 floats = 32 lanes x float4 (one 512B burst)
            float4 a = *(const float4*)(h_nodes + (size_t)im * DN + lane * 4);
            *(float4*)(row + lane * 4) = a;
            // h_nodes[jm]
            float4 b = *(const float4*)(h_nodes + (size_t)jm * DN + lane * 4);
            *(float4*)(row + DN + lane * 4) = b;
            // h_edges[im, jm]: 64 floats = 32 lanes x float2 (one 256B burst)
            float2 e = *(const float2*)(h_edges +
                ((size_t)im * N_NODES + (size_t)jm) * DE + lane * 2);
            *(float2*)(row + 2 * DN + lane * 2) = e;
        }
        // Same-wave LDS RAW: DS pipe is in-order, but fence the compiler and
        // drain DScnt explicitly before the WMMA read phase.
        asm volatile("s_wait_dscnt 0x0" ::: "memory");

        // 16x16 f32 accumulator, K = 320 in 80 steps of V_WMMA_F32_16X16X4_F32.
        // Inner loop is branch-free: 2x ds_load_b64 + 1x wmma per step.
        v8f acc = {0.f, 0.f, 0.f, 0.f, 0.f, 0.f, 0.f, 0.f};
        const float* aBase = featT + arow * KPAD + kOff;
        const float* bBase = ldsW + wRow * KPAD + kOff;
        #pragma unroll 4
        for (int kb = 0; kb < 80; ++kb) {
            const int k0 = kb * 4;
            v2f afrag = *(const v2f*)(aBase + k0);
            v2f bfrag = *(const v2f*)(bBase + k0);
            acc = __builtin_amdgcn_wmma_f32_16x16x4_f32(
                false, afrag, false, bfrag, (short)0, acc, false, false);
        }

        // Spill D to LDS: VGPR v, lanes 0-15 -> row M=v; lanes 16-31 -> row M=v+8.
        const int rowHi = (lane < 16) ? 0 : 8;
        #pragma unroll
        for (int v = 0; v < 8; ++v) {
            scoreT[(v + rowHi) * 16 + (lane & 15)] = acc[v];
        }
        asm volatile("s_wait_dscnt 0x0" ::: "memory");

        // Channel select + feasibility mask; coalesced 64B store per tile.
        if (lane < 16) {
            float s = scoreT[lane * 16 + bv] + ldsBias[bv];
            out[base + lane] = fv ? s : -__builtin_inff();
        }
    }
}

// ---------------------------------------------------------------------------
// STOP score: mean-pool h_nodes (1024x128, fully coalesced streaming read),
// dot with W_stop, add bias -> out[E]. One 128-thread block.
// ---------------------------------------------------------------------------
__global__ __launch_bounds__(128) void stop_score_kernel(
    const float* __restrict__ h_nodes,
    const float* __restrict__ W_stop,
    const float* __restrict__ b_stop,
    float* __restrict__ out)
{
    __shared__ float red[128];
    const int d = threadIdx.x;   // feature dim 0..127
    float s = 0.f;
    for (int n = 0; n < N_NODES; ++n) {
        s += h_nodes[(size_t)n * DN + d];   // consecutive lanes -> consecutive addrs
    }
    red[d] = (s * (1.0f / (float)N_NODES)) * W_stop[d];
    __syncthreads();
    for (int off = 64; off > 0; off >>= 1) {
        if (d < off) red[d] += red[d + off];
        __syncthreads();
    }
    if (d == 0) out[E_EDITS] = red[0] + b_stop[0];
}

// ---------------------------------------------------------------------------
extern "C" void kernel_launch(void* const* d_in, const int* in_sizes, int n_in,
                              void* d_out, int out_size, void* d_ws, size_t ws_size,
                              hipStream_t stream)
{
    (void)in_sizes; (void)n_in; (void)d_ws; (void)ws_size; (void)out_size;

    const float*   h_nodes  = (const float*)d_in[0];
    const float*   h_edges  = (const float*)d_in[1];
    const float*   W_edit   = (const float*)d_in[2];
    const float*   b_edit   = (const float*)d_in[3];
    const float*   W_stop   = (const float*)d_in[4];
    const float*   b_stop   = (const float*)d_in[5];
    const int*     i_idx    = (const int*)d_in[6];
    const int*     j_idx    = (const int*)d_in[7];
    const int*     b_idx    = (const int*)d_in[8];
    const uint8_t* feasible = (const uint8_t*)d_in[9];  // jnp bool_ = 1 byte/elem
    float*         out      = (float*)d_out;

    edit_scores_kernel<<<BLOCKS, 64, 0, stream>>>(
        h_nodes, h_edges, W_edit, b_edit, i_idx, j_idx, b_idx, feasible, out);
    stop_score_kernel<<<1, 128, 0, stream>>>(h_nodes, W_stop, b_stop, out);
}